// CTCLayer_4518305595673
// MI455X (gfx1250) — compile-verified
//
#include <hip/hip_runtime.h>

// CTC negative log-likelihood, blank = C-1, full-length inputs.
// B=4096, T=128, C=96, L=16 -> S=33. One wave32 per batch element:
// lane s holds alpha[s] (s=0..31), state 32 in an extra replicated register.
// y_pred rows streamed global->LDS with a 16-deep async-to-LDS pipeline
// (deep enough in-flight bytes chip-wide to saturate HBM at ~800ns latency).

#define NEGF (-1e30f)
#define EPSF (1e-7f)

constexpr int TT = 128;
constexpr int CC = 96;    // classes, blank = 95
constexpr int LL = 16;    // label length, S = 2*LL+1 = 33
constexpr int WAVES = 8;  // waves per block
constexpr int DEPTH = 16; // async pipeline depth (power of 2, <= TT)

#if defined(__has_builtin)
#  if __has_builtin(__builtin_amdgcn_global_load_async_to_lds_b128)
#    define HAVE_ASYNC_LDS 1
#  endif
#  if __has_builtin(__builtin_amdgcn_s_wait_asynccnt)
#    define HAVE_WAIT_ASYNC 1
#  endif
#endif

#if HAVE_WAIT_ASYNC
#  define WAIT_ASYNC(n) __builtin_amdgcn_s_wait_asynccnt(n)
#elif HAVE_ASYNC_LDS
#  define WAIT_ASYNC(n) asm volatile("s_wait_asynccnt " #n ::: "memory")
#else
#  define WAIT_ASYNC(n) __builtin_amdgcn_wave_barrier()
#endif

#if HAVE_ASYNC_LDS
typedef int v4i __attribute__((vector_size(4 * sizeof(int))));
typedef __attribute__((address_space(1))) v4i* gv4i_ptr;
typedef __attribute__((address_space(3))) v4i* lv4i_ptr;
#endif

// Copy one 96-float row (384 B) from global to LDS; lanes 0..23 move 16 B each.
__device__ __forceinline__ void stage_row(const float* g, float* l, int lane) {
#if HAVE_ASYNC_LDS
    if (lane < 24) {
        __builtin_amdgcn_global_load_async_to_lds_b128(
            (gv4i_ptr)(g + lane * 4),
            (lv4i_ptr)(l + lane * 4),
            0, 0);
    }
#else
    if (lane < 24) {
        const float4 v = ((const float4*)g)[lane];
        ((float4*)l)[lane] = v;
    }
    __builtin_amdgcn_wave_barrier();
#endif
}

__device__ __forceinline__ float lse3(float a, float b, float c) {
    float m = fmaxf(fmaxf(a, b), c);
    float s = __expf(a - m) + __expf(b - m) + __expf(c - m);
    return m + __logf(s);
}

__global__ __launch_bounds__(WAVES * 32)
void ctc_loss_kernel(const int* __restrict__ y_true,
                     const float* __restrict__ y_pred,
                     float* __restrict__ out, int B) {
    __shared__ float rows[WAVES][DEPTH][CC];

    const int lane = threadIdx.x & 31;
    const int w    = threadIdx.x >> 5;
    const int b    = blockIdx.x * WAVES + w;
    if (b >= B) return;  // whole wave exits together; no block barriers used

    const float* gp   = y_pred + (size_t)b * TT * CC;
    const int*   yt   = y_true + (size_t)b * LL;
    float*       base = &rows[w][0][0];

    // ext[s]: even s -> blank (CC-1); odd s -> label[(s-1)/2].
    // allow2[s]: only odd s>=3 with label[k] != label[k-1].
    int  ext_lane;
    bool allow2;
    if (lane & 1) {
        int k   = lane >> 1;
        int lab = yt[k];
        ext_lane = lab;
        allow2   = (lane >= 3) && (lab != yt[k - 1]);
    } else {
        ext_lane = CC - 1;
        allow2   = false;
    }

    // Fill the pipeline: rows 0..DEPTH-1 into slots 0..DEPTH-1.
#pragma unroll
    for (int r = 0; r < DEPTH; ++r)
        stage_row(gp + (size_t)r * CC, base + r * CC, lane);
    WAIT_ASYNC(15);  // in-order completion: row 0 has landed

    // alpha0: s=0 -> logp(blank), s=1 -> logp(label0), else NEG; alpha[32]=NEG.
    float lp0 = __logf(base[ext_lane] + EPSF);
    float A   = (lane <= 1) ? lp0 : NEGF;
    float A32 = NEGF;

    auto step = [&](int t) {
        const float* cur = base + (t & (DEPTH - 1)) * CC;

        float p   = cur[ext_lane];
        float lp  = __logf(p + EPSF);
        float lpb = __shfl(lp, 0, 32);     // lane 0 is blank -> lp for s=32

        float u1  = __shfl_up(A, 1, 32);
        float u2  = __shfl_up(A, 2, 32);
        float a31 = __shfl(A, 31, 32);

        float a1 = (lane >= 1) ? u1 : NEGF;
        float a2 = allow2 ? u2 : NEGF;

        float nA   = lse3(A, a1, a2) + lp;
        float nA32 = lse3(A32, a31, NEGF) + lpb;  // s=32: even -> allow2 false
        A   = nA;
        A32 = nA32;
        // Keep the next slot-overwrite (async write) from being scheduled
        // above this iteration's LDS gather.
        asm volatile("" ::: "memory");
    };

    // Main: keep DEPTH rows in flight. Iter t stages row t+DEPTH-1, then
    // waits asynccnt <= DEPTH-1 => row t complete (async loads finish in order).
    for (int t = 1; t <= TT - DEPTH; ++t) {
        int r = t + DEPTH - 1;
        stage_row(gp + (size_t)r * CC, base + (r & (DEPTH - 1)) * CC, lane);
        WAIT_ASYNC(15);
        step(t);
    }
    // Tail: nothing left to stage; drain.
    for (int t = TT - DEPTH + 1; t < TT; ++t) {
        WAIT_ASYNC(0);
        step(t);
    }

    // loss = -logaddexp(alpha[S-1], alpha[S-2])
    float aS1 = A32;                 // s = 32
    float aS2 = __shfl(A, 31, 32);   // s = 31
    float m   = fmaxf(aS1, aS2);
    float ll  = m + __logf(__expf(aS1 - m) + __expf(aS2 - m));
    if (lane == 0) out[b] = -ll;
}

extern "C" void kernel_launch(void* const* d_in, const int* in_sizes, int n_in,
                              void* d_out, int out_size, void* d_ws, size_t ws_size,
                              hipStream_t stream) {
    const int*   y_true = (const int*)d_in[0];
    const float* y_pred = (const float*)d_in[1];
    float*       out    = (float*)d_out;

    const int B = out_size;  // output is [B, 1] float32
    const int blocks = (B + WAVES - 1) / WAVES;
    hipLaunchKernelGGL(ctc_loss_kernel, dim3(blocks), dim3(WAVES * 32), 0, stream,
                       y_true, y_pred, out, B);
}